// IoUScore_61658550501694
// MI455X (gfx1250) — compile-verified
//
#include <hip/hip_runtime.h>
#include <math.h>

// Problem constants from the reference (B,C,H,W) = (64,1,512,512)
#define B_BATCH 64
#define N_PER   262144              // C*H*W
#define NBINS   256
#define THREADS 256                 // 8 wave32s per block
#define TILES   8                   // float4-tiles per block per pass
#define ELEMS_PER_BLOCK (THREADS * 4 * TILES)        // 8192
#define BLOCKS_PER_BATCH (N_PER / ELEMS_PER_BLOCK)   // 32

// gfx1250 async load-to-LDS path (guarded so the file always compiles)
#if defined(__HIP_DEVICE_COMPILE__) && defined(__gfx1250__) && \
    __has_builtin(__builtin_amdgcn_global_load_async_to_lds_b128) && \
    __has_builtin(__builtin_amdgcn_s_wait_asynccnt)
#define USE_ASYNC_LDS 1
#else
#define USE_ASYNC_LDS 0
#endif

#if USE_ASYNC_LDS
typedef int v4i __attribute__((vector_size(16)));
typedef __attribute__((address_space(1))) v4i* gv4i_ptr;
typedef __attribute__((address_space(3))) v4i* lv4i_ptr;
#endif

__device__ __forceinline__ float sigm(float x) {
    return 1.0f / (1.0f + expf(-x));
}

// Order-preserving float <-> uint key (total order, handles negatives).
// Init values: min-key memset 0xFF (== UINT_MAX), max-key memset 0x00.
__device__ __forceinline__ unsigned int fkey(float f) {
    unsigned int u = __float_as_uint(f);
    return u ^ (((unsigned int)((int)u >> 31)) | 0x80000000u);
}
__device__ __forceinline__ float fkey_inv(unsigned int k) {
    unsigned int mask = ((unsigned int)((int)(~k) >> 31)) | 0x80000000u;
    return __uint_as_float(k ^ mask);
}

// ---------------------------------------------------------------- pass 1
// Per-batch min/max of RAW logits (sigmoid is monotone; applied later to
// the two scalars only). No transcendentals in this pass.
__global__ void __launch_bounds__(THREADS)
minmax_kernel(const float* __restrict__ outp,
              unsigned int* __restrict__ minKey,
              unsigned int* __restrict__ maxKey) {
    const int b   = blockIdx.y;
    const int tid = threadIdx.x;
    const size_t base = (size_t)b * N_PER + (size_t)blockIdx.x * ELEMS_PER_BLOCK;
    const float4* src = (const float4*)(outp + base);

    float lmin =  3.402823466e38f;
    float lmax = -3.402823466e38f;
#pragma unroll
    for (int t = 0; t < TILES; ++t) {
        float4 v = src[t * THREADS + tid];
        lmin = fminf(fminf(fminf(lmin, v.x), v.y), fminf(v.z, v.w));
        lmax = fmaxf(fmaxf(fmaxf(lmax, v.x), v.y), fmaxf(v.z, v.w));
    }

    __shared__ float smin[THREADS], smax[THREADS];
    smin[tid] = lmin; smax[tid] = lmax;
    __syncthreads();
    for (int o = THREADS / 2; o > 0; o >>= 1) {
        if (tid < o) {
            smin[tid] = fminf(smin[tid], smin[tid + o]);
            smax[tid] = fmaxf(smax[tid], smax[tid + o]);
        }
        __syncthreads();
    }
    if (tid == 0) {
        atomicMin(&minKey[b], fkey(smin[0]));
        atomicMax(&maxKey[b], fkey(smax[0]));
    }
}

// ---------------------------------------------------------------- pass 2
// Per-block LDS histogram (ds_add_u32), one global atomic per bin to flush.
__global__ void __launch_bounds__(THREADS)
hist_kernel(const float* __restrict__ outp,
            const unsigned int* __restrict__ minKey,
            const unsigned int* __restrict__ maxKey,
            unsigned int* __restrict__ ghist) {
    const int b   = blockIdx.y;
    const int tid = threadIdx.x;

    __shared__ unsigned int lh[NBINS];
    lh[tid] = 0u;
    __syncthreads();

    // pmin/pmax of the sigmoid stream == sigm(min/max of raw stream)
    const float pmin = sigm(fkey_inv(minKey[b]));
    const float pmax = sigm(fkey_inv(maxKey[b]));
    const float span = pmax - pmin;
    const float scale = (span > 0.0f) ? ((float)NBINS / span) : 0.0f;

    const size_t base = (size_t)b * N_PER + (size_t)blockIdx.x * ELEMS_PER_BLOCK;
    const float4* src = (const float4*)(outp + base);

#pragma unroll
    for (int t = 0; t < TILES; ++t) {
        float4 v = src[t * THREADS + tid];
        int i0 = min(NBINS - 1, max(0, (int)((sigm(v.x) - pmin) * scale)));
        int i1 = min(NBINS - 1, max(0, (int)((sigm(v.y) - pmin) * scale)));
        int i2 = min(NBINS - 1, max(0, (int)((sigm(v.z) - pmin) * scale)));
        int i3 = min(NBINS - 1, max(0, (int)((sigm(v.w) - pmin) * scale)));
        atomicAdd(&lh[i0], 1u);   // ds_add_u32
        atomicAdd(&lh[i1], 1u);
        atomicAdd(&lh[i2], 1u);
        atomicAdd(&lh[i3], 1u);
    }
    __syncthreads();
    atomicAdd(&ghist[b * NBINS + tid], lh[tid]);
}

// ---------------------------------------------------------------- pass 3 (tiny)
// Otsu via single running-cumsum pass:
//   w2[i+1] = N - w1[i],  m2[i+1] = (sumHC - cumHC[i]) / (N - w1[i])
// Emits the threshold in the RAW logit domain: xth = logit(centers[argmax]),
// so the big binarize pass needs only one compare per element.
__global__ void otsu_kernel(const unsigned int* __restrict__ ghist,
                            const unsigned int* __restrict__ minKey,
                            const unsigned int* __restrict__ maxKey,
                            float* __restrict__ thrX) {
    const int b = threadIdx.x;
    if (b >= B_BATCH) return;

    const float pmin = sigm(fkey_inv(minKey[b]));
    const float pmax = sigm(fkey_inv(maxKey[b]));
    const float span = pmax - pmin;
    const float binw = span / (float)NBINS;
    const unsigned int* h = ghist + b * NBINS;

    float sumHC = 0.0f;
    for (int i = 0; i < NBINS; ++i) {
        float c = pmin + ((float)i + 0.5f) * binw;
        sumHC += (float)h[i] * c;
    }

    const float Ntot = (float)N_PER;
    float w1 = 0.0f, cumHC = 0.0f, best = -3.402823466e38f;
    int bestI = 0;
    for (int i = 0; i < NBINS - 1; ++i) {
        float hi = (float)h[i];
        float c  = pmin + ((float)i + 0.5f) * binw;
        w1    += hi;
        cumHC += hi * c;
        float w2n = Ntot - w1;                 // == w2[i+1]
        if (w1 > 0.0f && w2n > 0.0f) {
            float m1 = cumHC / w1;             // m1[i]
            float m2 = (sumHC - cumHC) / w2n;  // m2[i+1]
            float d  = m1 - m2;
            float v  = w1 * w2n * d * d;
            if (v > best) { best = v; bestI = i; }  // first-max like argmax
        }
    }
    const float th = pmin + ((float)bestI + 0.5f) * binw;   // p-domain threshold
    thrX[b] = logf(th / (1.0f - th));                       // logit -> raw domain
}

// ---------------------------------------------------------------- pass 4
__device__ __forceinline__ void accum4(float4 ov, float4 tv, float xth,
                                       unsigned int& ci, unsigned int& cb,
                                       unsigned int& ct) {
    // sigm(x) > th  <=>  x > logit(th): one v_cmp per element, no exp/rcp.
    int b0 = ov.x > xth, b1 = ov.y > xth, b2 = ov.z > xth, b3 = ov.w > xth;
    int t0 = tv.x != 0.0f, t1 = tv.y != 0.0f;
    int t2 = tv.z != 0.0f, t3 = tv.w != 0.0f;
    ci += (unsigned)((b0 & t0) + (b1 & t1) + (b2 & t2) + (b3 & t3));
    cb += (unsigned)(b0 + b1 + b2 + b3);
    ct += (unsigned)(t0 + t1 + t2 + t3);
}

__global__ void __launch_bounds__(THREADS)
count_kernel(const float* __restrict__ outp, const float* __restrict__ tgt,
             const float* __restrict__ thrX,
             unsigned int* __restrict__ gInter,
             unsigned int* __restrict__ gSumB,
             unsigned int* __restrict__ gSumT) {
    const int b   = blockIdx.y;
    const int tid = threadIdx.x;
    const size_t base = (size_t)b * N_PER + (size_t)blockIdx.x * ELEMS_PER_BLOCK;
    const float4* osrc = (const float4*)(outp + base);
    const float4* tsrc = (const float4*)(tgt + base);
    const float xth = thrX[b];

    unsigned int ci = 0, cb = 0, ct = 0;

#if USE_ASYNC_LDS
    // Double-buffered async pipeline: target stream staged through LDS with
    // the CDNA5 async copy engine; each lane consumes only its own 16B slot,
    // so only s_wait_asynccnt (no barrier) is needed between stage & use.
    __shared__ __align__(16) float tbuf[2][THREADS * 4];
    auto issue = [&](int buf, int t) {
        __builtin_amdgcn_global_load_async_to_lds_b128(
            (gv4i_ptr)(tsrc + t * THREADS + tid),
            (lv4i_ptr)(&tbuf[buf][tid * 4]),
            0, 0);
    };
    issue(0, 0);
#pragma unroll
    for (int t = 0; t < TILES - 1; ++t) {
        issue((t + 1) & 1, t + 1);
        __builtin_amdgcn_s_wait_asynccnt(1);   // tile t complete (in-order)
        float4 ov = osrc[t * THREADS + tid];
        float4 tv = *(const float4*)(&tbuf[t & 1][tid * 4]);
        accum4(ov, tv, xth, ci, cb, ct);
    }
    __builtin_amdgcn_s_wait_asynccnt(0);
    {
        const int t = TILES - 1;
        float4 ov = osrc[t * THREADS + tid];
        float4 tv = *(const float4*)(&tbuf[t & 1][tid * 4]);
        accum4(ov, tv, xth, ci, cb, ct);
    }
#else
#pragma unroll
    for (int t = 0; t < TILES; ++t) {
        float4 ov = osrc[t * THREADS + tid];
        float4 tv = tsrc[t * THREADS + tid];
        accum4(ov, tv, xth, ci, cb, ct);
    }
#endif

    __shared__ unsigned int sI[THREADS], sB[THREADS], sT[THREADS];
    sI[tid] = ci; sB[tid] = cb; sT[tid] = ct;
    __syncthreads();
    for (int o = THREADS / 2; o > 0; o >>= 1) {
        if (tid < o) {
            sI[tid] += sI[tid + o];
            sB[tid] += sB[tid + o];
            sT[tid] += sT[tid + o];
        }
        __syncthreads();
    }
    if (tid == 0) {
        atomicAdd(&gInter[b], sI[0]);
        atomicAdd(&gSumB[b],  sB[0]);
        atomicAdd(&gSumT[b],  sT[0]);
    }
}

// ---------------------------------------------------------------- pass 5 (tiny)
__global__ void finalize_kernel(const unsigned int* __restrict__ gInter,
                                const unsigned int* __restrict__ gSumB,
                                const unsigned int* __restrict__ gSumT,
                                float* __restrict__ out) {
    __shared__ float s[B_BATCH];
    const int b = threadIdx.x;
    if (b < B_BATCH) {
        unsigned int I = gInter[b];
        unsigned int U = gSumB[b] + gSumT[b] - I;
        s[b] = ((float)I + 1.0f) / ((float)U + 1.0f);
    }
    __syncthreads();
    for (int o = B_BATCH / 2; o > 0; o >>= 1) {
        if (b < o) s[b] += s[b + o];
        __syncthreads();
    }
    if (b == 0) out[0] = s[0] / (float)B_BATCH;
}

// ---------------------------------------------------------------- host glue
extern "C" void kernel_launch(void* const* d_in, const int* in_sizes, int n_in,
                              void* d_out, int out_size, void* d_ws, size_t ws_size,
                              hipStream_t stream) {
    const float* outp = (const float*)d_in[0];
    const float* tgt  = (const float*)d_in[1];
    float* res        = (float*)d_out;

    // workspace layout (bytes)
    char* ws = (char*)d_ws;
    unsigned int* minKey  = (unsigned int*)(ws + 0);       //   64 * 4
    unsigned int* maxKey  = (unsigned int*)(ws + 256);     //   64 * 4
    unsigned int* ghist   = (unsigned int*)(ws + 512);     // 64*256*4 = 65536
    float*        thrX    = (float*)       (ws + 66048);   //   64 * 4
    unsigned int* vInter  = (unsigned int*)(ws + 66304);   //   64 * 4
    unsigned int* vSumB   = (unsigned int*)(ws + 66560);   //   64 * 4
    unsigned int* vSumT   = (unsigned int*)(ws + 66816);   //   64 * 4
    // total used: 67072 bytes

    // min-keys -> 0xFFFFFFFF (UINT_MAX); everything else -> 0
    (void)hipMemsetAsync(minKey, 0xFF, 256, stream);
    (void)hipMemsetAsync(maxKey, 0x00, 67072 - 256, stream);

    dim3 grid(BLOCKS_PER_BATCH, B_BATCH);
    minmax_kernel<<<grid, THREADS, 0, stream>>>(outp, minKey, maxKey);
    hist_kernel<<<grid, THREADS, 0, stream>>>(outp, minKey, maxKey, ghist);
    otsu_kernel<<<1, B_BATCH, 0, stream>>>(ghist, minKey, maxKey, thrX);
    count_kernel<<<grid, THREADS, 0, stream>>>(outp, tgt, thrX, vInter, vSumB, vSumT);
    finalize_kernel<<<1, B_BATCH, 0, stream>>>(vInter, vSumB, vSumT, res);
}